// CustomSparseAttention_60318520705504
// MI455X (gfx1250) — compile-verified
//
#include <hip/hip_runtime.h>

typedef __attribute__((ext_vector_type(2))) float v2f;
typedef __attribute__((ext_vector_type(4))) float v4f;
typedef __attribute__((ext_vector_type(8))) float v8f;

#define EMBED 1024
#define T_LEN 2048
#define S_LEN 2048
#define NHEAD 16
#define HDIM  64
#define LDSW  68   // padded row stride (floats): 16B-aligned rows, conflict-free b64 reads

__device__ __forceinline__ v8f wmma4(v2f a, v2f b, v8f c) {
  // D = A(16x4 f32) * B(4x16 f32) + C(16x16 f32), exact fp32
  return __builtin_amdgcn_wmma_f32_16x16x4_f32(false, a, false, b, (short)0, c,
                                               false, false);
}

// ---- CDNA5 async global->LDS copy (ASYNCcnt) with safe fallback ----------
#if defined(__has_builtin)
#if __has_builtin(__builtin_amdgcn_global_load_async_to_lds_b128) && \
    __has_builtin(__builtin_amdgcn_s_wait_asynccnt)
#define HAVE_ASYNC_LDS 1
#endif
#endif

// Builtin signature (leaked by round-2 diagnostic): param0 is a non-const
// pointer to a 16-byte int vector in the global AS; param1 mirrors it in LDS.
typedef int v4i_t __attribute__((vector_size(16)));
typedef __attribute__((address_space(1))) v4i_t* gas_v4i_ptr;
typedef __attribute__((address_space(3))) v4i_t* las_v4i_ptr;

__device__ __forceinline__ void copy16_async(const float* g, float* l) {
#ifdef HAVE_ASYNC_LDS
  __builtin_amdgcn_global_load_async_to_lds_b128((gas_v4i_ptr)g,
                                                 (las_v4i_ptr)l, 0, 0);
#else
  *(v4f*)l = *(const v4f*)g;
#endif
}

__device__ __forceinline__ void wait_async_copies() {
#ifdef HAVE_ASYNC_LDS
  __builtin_amdgcn_s_wait_asynccnt(0);
#endif
}

// ---------------------------------------------------------------------------
// Kernel 1: Y[M,1024] = X[M,1024] @ W^T + bias, optionally scaled by
// 0.125 * sigmoid(temperature).  Block = 4 waves; block computes a 64x64
// output tile.  X/W 64x64 K-tiles are staged in LDS via async b128 copies
// (one copy per block instead of 4x duplicated strided global reads), then
// fed to V_WMMA_F32_16X16X4_F32 from conflict-free padded LDS rows.
// ---------------------------------------------------------------------------
__global__ void __launch_bounds__(128)
gemm_xwT_kernel(const float* __restrict__ X, const float* __restrict__ W,
                const float* __restrict__ bias, float* __restrict__ Y,
                const float* __restrict__ temperature) {
  __shared__ float lx[64 * LDSW];
  __shared__ float lw[64 * LDSW];

  const int tid  = threadIdx.x;
  const int lane = tid & 31;
  const int wave = tid >> 5;
  const int mb = blockIdx.x * 64;       // block row base
  const int n0 = blockIdx.y * 64;       // block col base
  const int r  = lane & 15;             // A row / B col within 16-tile
  const int hi = lane >> 4;             // upper half-wave -> K = 2,3

  float scale = 1.0f;
  if (temperature) {
    float t = temperature[0];
    scale = 0.125f * (1.0f / (1.0f + __expf(-t)));  // D^-1/2 * sigmoid(T)
  }

  const int crow = tid >> 4;            // 0..7   staging row within group
  const int ccol = (tid & 15) * 4;      // 0..60  staging float4 column

  v8f a0 = {}, a1 = {}, a2 = {}, a3 = {};
  for (int k0 = 0; k0 < EMBED; k0 += 64) {
    // ---- stage 64x64 tiles of X and W into LDS (async, b128/lane) ----
#pragma unroll
    for (int i = 0; i < 8; ++i) {
      const int row = i * 8 + crow;
      copy16_async(X + (size_t)(mb + row) * EMBED + k0 + ccol,
                   &lx[row * LDSW + ccol]);
      copy16_async(W + (size_t)(n0 + row) * EMBED + k0 + ccol,
                   &lw[row * LDSW + ccol]);
    }
    if (k0 + 64 < EMBED) {  // keep next tile's lines inbound in L2
      __builtin_prefetch(X + (size_t)(mb + crow) * EMBED + k0 + 64 + ccol, 0, 1);
      __builtin_prefetch(W + (size_t)(n0 + crow) * EMBED + k0 + 64 + ccol, 0, 1);
    }
    wait_async_copies();
    __syncthreads();

    const float* ax = &lx[(wave * 16 + r) * LDSW + hi * 2];
    const float* b0p = &lw[(r)      * LDSW + hi * 2];
    const float* b1p = &lw[(16 + r) * LDSW + hi * 2];
    const float* b2p = &lw[(32 + r) * LDSW + hi * 2];
    const float* b3p = &lw[(48 + r) * LDSW + hi * 2];
#pragma unroll
    for (int k = 0; k < 64; k += 4) {
      v2f av = *(const v2f*)(ax + k);
      a0 = wmma4(av, *(const v2f*)(b0p + k), a0);
      a1 = wmma4(av, *(const v2f*)(b1p + k), a1);
      a2 = wmma4(av, *(const v2f*)(b2p + k), a2);
      a3 = wmma4(av, *(const v2f*)(b3p + k), a3);
    }
    __syncthreads();
  }

#pragma unroll
  for (int i = 0; i < 8; ++i) {
    const int m = mb + wave * 16 + i + 8 * hi;
    float* yr = Y + (size_t)m * EMBED + n0 + r;
    yr[0]  = (a0[i] + bias[n0 + r])      * scale;
    yr[16] = (a1[i] + bias[n0 + 16 + r]) * scale;
    yr[32] = (a2[i] + bias[n0 + 32 + r]) * scale;
    yr[48] = (a3[i] + bias[n0 + 48 + r]) * scale;
  }
}

// ---------------------------------------------------------------------------
// Kernel 2: fused masked attention for one (b, h, 16-row query tile).
// 8 waves / 256 threads.  LDS: 16x2048 score block (128 KB) + 32 KB partials
// -> 160 KB dynamic, 2 workgroups per WGP (320 KB LDS).
// Phase 1: scores = QK^T via WMMA, mask applied, -> LDS
// Phase 2: wave32 shuffle softmax per row (float4 LDS passes); attn written
//          exactly once with non-temporal b128 stores (537 MB HBM stream)
// Phase 3: O = P @ V via WMMA over per-wave S-slices, LDS tree reduce
// ---------------------------------------------------------------------------
__global__ void __launch_bounds__(256)
sparse_attn_kernel(const float* __restrict__ q, const float* __restrict__ kmat,
                   const float* __restrict__ vmat,
                   const unsigned char* __restrict__ mask,
                   float* __restrict__ attn, float* __restrict__ oheads) {
  extern __shared__ float smem[];
  float* sc   = smem;                 // 16 x 2048 scores / probabilities
  float* part = smem + 16 * S_LEN;    // [8 waves][4 dt][8][32] partials

  const int lane = threadIdx.x & 31;
  const int wave = threadIdx.x >> 5;
  const int r  = lane & 15;
  const int hi = lane >> 4;
  const int t0 = blockIdx.x * 16;
  const int h  = blockIdx.y;
  const int b  = blockIdx.z;

  const float* qrow =
      q + (size_t)(b * T_LEN + t0 + r) * EMBED + h * HDIM + hi * 2;

  // ---- Phase 1: masked scores into LDS (each wave: 256 columns of S) ----
  for (int st = 0; st < 16; ++st) {
    const int s0 = wave * 256 + st * 16;
    const float* krow =
        kmat + (size_t)(b * S_LEN + s0 + r) * EMBED + h * HDIM + hi * 2;
    if (st + 1 < 16)  // prefetch next K tile row for this lane
      __builtin_prefetch(krow + 16 * EMBED, 0, 1);
    v8f c = {};
#pragma unroll
    for (int kk = 0; kk < HDIM; kk += 4) {
      v2f av = *(const v2f*)(qrow + kk);
      v2f bv = *(const v2f*)(krow + kk);
      c = wmma4(av, bv, c);
    }
    const int n = s0 + r;
#pragma unroll
    for (int i = 0; i < 8; ++i) {
      const int m = i + 8 * hi;
      const bool keep = mask[(size_t)(t0 + m) * S_LEN + n] != 0;
      sc[m * S_LEN + n] = keep ? c[i] : -__builtin_inff();
    }
  }
  __syncthreads();

  // ---- Phase 2: row softmax (each wave: 2 rows), stream attn out ----
#pragma unroll
  for (int rr = 0; rr < 2; ++rr) {
    const int row = wave * 2 + rr;
    float* srow = sc + row * S_LEN;
    float mx = -__builtin_inff();
    for (int c = lane * 4; c < S_LEN; c += 128) {
      v4f x = *(const v4f*)(srow + c);
      mx = fmaxf(mx, fmaxf(fmaxf(x[0], x[1]), fmaxf(x[2], x[3])));
    }
#pragma unroll
    for (int off = 16; off > 0; off >>= 1) mx = fmaxf(mx, __shfl_xor(mx, off, 32));
    float sum = 0.0f;
    for (int c = lane * 4; c < S_LEN; c += 128) {
      v4f x = *(const v4f*)(srow + c);
      v4f p;
#pragma unroll
      for (int j = 0; j < 4; ++j) p[j] = __expf(x[j] - mx);  // exp(-inf)=0
      *(v4f*)(srow + c) = p;
      sum += (p[0] + p[1]) + (p[2] + p[3]);
    }
#pragma unroll
    for (int off = 16; off > 0; off >>= 1) sum += __shfl_xor(sum, off, 32);
    const float inv = 1.0f / sum;
    float* arow =
        attn + ((size_t)(b * NHEAD + h) * T_LEN + t0 + row) * S_LEN;
    for (int c = lane * 4; c < S_LEN; c += 128) {
      v4f p = *(const v4f*)(srow + c);
#pragma unroll
      for (int j = 0; j < 4; ++j) p[j] *= inv;
      *(v4f*)(srow + c) = p;
      __builtin_nontemporal_store(p, (v4f*)(arow + c));  // attn never re-read
    }
  }
  __syncthreads();

  // ---- Phase 3: O(16x64) = P(16x2048) @ V(2048x64), K sliced by wave ----
  v8f o0 = {}, o1 = {}, o2 = {}, o3 = {};
  const int sbase = wave * 256;
  const float* vhead = vmat + (size_t)b * S_LEN * EMBED + h * HDIM;
  for (int kk = 0; kk < 256; kk += 4) {
    const int kb = sbase + kk + hi * 2;
    v2f av = *(const v2f*)(sc + r * S_LEN + kb);
    const float* v0 = vhead + (size_t)kb * EMBED + r;
    const float* v1 = v0 + EMBED;
    if (kk + 4 < 256)
      __builtin_prefetch(v0 + 4 * EMBED, 0, 1);
    v2f b0 = {v0[0],  v1[0]};
    v2f b1 = {v0[16], v1[16]};
    v2f b2 = {v0[32], v1[32]};
    v2f b3 = {v0[48], v1[48]};
    o0 = wmma4(av, b0, o0);
    o1 = wmma4(av, b1, o1);
    o2 = wmma4(av, b2, o2);
    o3 = wmma4(av, b3, o3);
  }
#pragma unroll
  for (int i = 0; i < 8; ++i) {
    part[((wave * 4 + 0) * 8 + i) * 32 + lane] = o0[i];
    part[((wave * 4 + 1) * 8 + i) * 32 + lane] = o1[i];
    part[((wave * 4 + 2) * 8 + i) * 32 + lane] = o2[i];
    part[((wave * 4 + 3) * 8 + i) * 32 + lane] = o3[i];
  }
  __syncthreads();

  if (wave < 4) {  // waves 0..3 each reduce one 16-wide d-tile
    const int dt = wave;
#pragma unroll
    for (int i = 0; i < 8; ++i) {
      float s = 0.0f;
#pragma unroll
      for (int w = 0; w < 8; ++w)
        s += part[((w * 4 + dt) * 8 + i) * 32 + lane];
      const int m = i + 8 * hi;
      oheads[(size_t)(b * T_LEN + t0 + m) * EMBED + h * HDIM + dt * 16 + r] = s;
    }
  }
}

// ---------------------------------------------------------------------------
extern "C" void kernel_launch(void* const* d_in, const int* in_sizes, int n_in,
                              void* d_out, int out_size, void* d_ws,
                              size_t ws_size, hipStream_t stream) {
  (void)in_sizes; (void)n_in; (void)out_size; (void)ws_size;
  const float* query = (const float*)d_in[0];
  const float* key   = (const float*)d_in[1];
  const float* value = (const float*)d_in[2];
  const float* Wq = (const float*)d_in[3];
  const float* bq = (const float*)d_in[4];
  const float* Wk = (const float*)d_in[5];
  const float* bk = (const float*)d_in[6];
  const float* Wv = (const float*)d_in[7];
  const float* bv = (const float*)d_in[8];
  const float* Wo = (const float*)d_in[9];
  const float* bo = (const float*)d_in[10];
  const float* temperature = (const float*)d_in[11];
  const unsigned char* mask = (const unsigned char*)d_in[12];

  const size_t MAT = (size_t)2 * T_LEN * EMBED;  // 4,194,304 elements
  float* qws = (float*)d_ws;       // scaled Q projection
  float* kws = qws + MAT;          // K projection
  float* vws = kws + MAT;          // V projection
  float* ohs = vws + MAT;          // merged head outputs [B,T,E]

  float* out  = (float*)d_out;     // [B,T,E]
  float* attn = out + MAT;         // [B,H,T,S]

  dim3 gGrid(64, 16, 1);           // M/64 x 1024/64
  dim3 gBlk(128, 1, 1);
  gemm_xwT_kernel<<<gGrid, gBlk, 0, stream>>>(query, Wq, bq, qws, temperature);
  gemm_xwT_kernel<<<gGrid, gBlk, 0, stream>>>(key,   Wk, bk, kws, nullptr);
  gemm_xwT_kernel<<<gGrid, gBlk, 0, stream>>>(value, Wv, bv, vws, nullptr);

  dim3 aGrid(T_LEN / 16, NHEAD, 2);
  dim3 aBlk(256, 1, 1);
  const size_t shbytes = (16 * S_LEN + 8 * 4 * 8 * 32) * sizeof(float);  // 160 KB
  sparse_attn_kernel<<<aGrid, aBlk, shbytes, stream>>>(qws, kws, vws, mask,
                                                       attn, ohs);

  gemm_xwT_kernel<<<gGrid, gBlk, 0, stream>>>(ohs, Wo, bo, out, nullptr);
}